// SO3net_28432683499863
// MI455X (gfx1250) — compile-verified
//
#include <hip/hip_runtime.h>
#include <hip/hip_bf16.h>

// ============================================================================
// SO3net forward for MI455X (gfx1250), fp32 end-to-end with V_WMMA_F32_16X16X4.
// One workgroup (8 wave32) per batch element; all state LDS-resident (~149KB).
// ============================================================================

#define BATCHN 1024
#define NA 12
#define NE 66
#define CCH 128
#define NR 20

typedef float v2f __attribute__((ext_vector_type(2)));
typedef float v8f __attribute__((ext_vector_type(8)));

struct KParams {
    const float* pos;        // [B,12,3]
    const int*   zed;        // [B,12]
    const float* embed;      // [6,128]
    const float* rad[23];    // each [20,128]
    const float* lin[9];     // [128,128]x6, [128,68],[128,36],[128,4]
    const float* cg;         // 245 floats in d_ws
    float*       out;        // [B,196]
};

// ---------------------------------------------------------------------------
// CG offsets inside the packed 245-float table
// combos: (0,0,0)@0(1) (0,1,1)@1(9) (1,0,1)@10(9) (1,1,0)@19(9) (1,1,1)@28(27)
//         (1,2,1)@55(45) (0,2,2)@100(25) (1,1,2)@125(45) (1,2,2)@170(75)
// ---------------------------------------------------------------------------
__device__ __forceinline__ int cgoff(int l1, int l2, int l3) {
    int code = l1 * 9 + l2 * 3 + l3;
    switch (code) {
        case 0:  return 0;
        case 4:  return 1;
        case 10: return 10;
        case 12: return 19;
        case 13: return 28;
        case 16: return 55;
        case 8:  return 100;
        case 14: return 125;
        default: return 170;   // code 17 -> (1,2,2)
    }
}

// ---------------------------------------------------------------------------
// Device-side e3nn real Clebsch-Gordan table builder (double precision)
// ---------------------------------------------------------------------------
struct cplx { double re, im; };
__device__ __forceinline__ cplx cmul(cplx a, cplx b) { return {a.re*b.re - a.im*b.im, a.re*b.im + a.im*b.re}; }
__device__ __forceinline__ cplx cadd(cplx a, cplx b) { return {a.re + b.re, a.im + b.im}; }
__device__ __forceinline__ cplx cconj(cplx a)        { return {a.re, -a.im}; }
__device__ __forceinline__ cplx cscale(cplx a, double s) { return {a.re*s, a.im*s}; }

__device__ double factd(int n) { double r = 1.0; for (int i = 2; i <= n; ++i) r *= (double)i; return r; }

__device__ double su2cg(int j1, int j2, int j3, int m1, int m2, int m3) {
    if (m1 + m2 != m3) return 0.0;
    double pref = sqrt((2.0*j3 + 1.0) * factd(j3+j1-j2) * factd(j3-j1+j2) * factd(j1+j2-j3) / factd(j1+j2+j3+1));
    pref *= sqrt(factd(j3+m3)*factd(j3-m3)*factd(j1-m1)*factd(j1+m1)*factd(j2-m2)*factd(j2+m2));
    int kmin = 0;
    if (-(j3 - j2 + m1) > kmin) kmin = -(j3 - j2 + m1);
    if (-(j3 - j1 - m2) > kmin) kmin = -(j3 - j1 - m2);
    int kmax = j1 + j2 - j3;
    if (j1 - m1 < kmax) kmax = j1 - m1;
    if (j2 + m2 < kmax) kmax = j2 + m2;
    double s = 0.0;
    for (int k = kmin; k <= kmax; ++k) {
        double den = factd(k)*factd(j1+j2-j3-k)*factd(j1-m1-k)*factd(j2+m2-k)*factd(j3-j2+m1+k)*factd(j3-j1-m2+k);
        s += ((k & 1) ? -1.0 : 1.0) / den;
    }
    return pref * s;
}

__device__ void buildq(int l, cplx q[5][5]) {
    for (int a = 0; a < 5; ++a) for (int b = 0; b < 5; ++b) q[a][b] = {0.0, 0.0};
    const double inv = 0.70710678118654752440;
    for (int m = -l; m < 0; ++m) {
        q[l + m][l - m].re = inv;     // col l+|m|
        q[l + m][l + m].im = -inv;    // col l-|m|
    }
    q[l][l].re = 1.0;
    for (int m = 1; m <= l; ++m) {
        double sg = (m & 1) ? -1.0 : 1.0;
        q[l + m][l + m].re = sg * inv;
        q[l + m][l - m].im = sg * inv;
    }
    cplx f;
    if (l == 0)      f = {1.0, 0.0};
    else if (l == 1) f = {0.0, -1.0};   // (-i)^1
    else             f = {-1.0, 0.0};   // (-i)^2
    for (int a = 0; a < 5; ++a) for (int b = 0; b < 5; ++b) q[a][b] = cmul(f, q[a][b]);
}

__global__ void so3_cg_init(float* cg) {
    const int combos[9][4] = {
        {0,0,0,0},{0,1,1,1},{1,0,1,10},{1,1,0,19},{1,1,1,28},
        {1,2,1,55},{0,2,2,100},{1,1,2,125},{1,2,2,170}
    };
    int t = threadIdx.x;
    if (t >= 9) return;
    int l1 = combos[t][0], l2 = combos[t][1], l3 = combos[t][2], off = combos[t][3];
    int d1 = 2*l1 + 1, d2 = 2*l2 + 1, d3 = 2*l3 + 1;
    cplx q1[5][5], q2[5][5], q3[5][5];
    buildq(l1, q1); buildq(l2, q2); buildq(l3, q3);
    double re[75], im[75];
    double maxre = 0.0, maxim = 0.0;
    for (int a = 0; a < d1; ++a)
        for (int b = 0; b < d2; ++b)
            for (int c = 0; c < d3; ++c) {
                cplx s = {0.0, 0.0};
                for (int m = 0; m < d1; ++m)
                    for (int n = 0; n < d2; ++n) {
                        int p = (m - l1) + (n - l2) + l3;     // m3 fixed by m1+m2
                        if (p < 0 || p >= d3) continue;
                        double w = su2cg(l1, l2, l3, m - l1, n - l2, (m - l1) + (n - l2));
                        if (w == 0.0) continue;
                        cplx term = cmul(cmul(q1[m][a], q2[n][b]), cconj(q3[p][c]));
                        s = cadd(s, cscale(term, w));
                    }
                int idx = (a * d2 + b) * d3 + c;
                re[idx] = s.re; im[idx] = s.im;
                if (fabs(s.re) > maxre) maxre = fabs(s.re);
                if (fabs(s.im) > maxim) maxim = fabs(s.im);
            }
    bool useRe = (maxre >= maxim);
    for (int i = 0; i < d1 * d2 * d3; ++i) cg[off + i] = (float)(useRe ? re[i] : im[i]);
}

// ---------------------------------------------------------------------------
// Layer / path bookkeeping (insertion-order path enumeration of the reference)
// ---------------------------------------------------------------------------
struct PathT { int l1, l2, rad; };
struct BlkT  { int l3, npaths, lin, resid; PathT p[4]; };

__constant__ BlkT BLKS[9] = {
    {0, 1, 0, 1, {{0,0,0}}},                                      // L0 l3=0
    {1, 1, 1, 0, {{0,1,1}}},                                      // L0 l3=1
    {0, 2, 2, 1, {{0,0,2},{1,1,3}}},                              // L1 l3=0
    {1, 4, 3, 1, {{0,1,4},{1,0,5},{1,1,6},{1,2,7}}},              // L1 l3=1
    {0, 2, 4, 1, {{0,0,8},{1,1,9}}},                              // L2 l3=0
    {1, 4, 5, 1, {{0,1,10},{1,0,11},{1,1,12},{1,2,13}}},          // L2 l3=1
    {0, 2, 6, 0, {{0,0,14},{1,1,15}}},                            // L3 l3=0
    {1, 4, 7, 0, {{0,1,16},{1,0,17},{1,1,18},{1,2,19}}},          // L3 l3=1
    {2, 3, 8, 0, {{0,2,20},{1,1,21},{1,2,22}}},                   // L3 l3=2
};

#define ACW 130   // padded channel stride for s_acc (bank-conflict free A loads)

__global__ __launch_bounds__(256) void so3_main(KParams P) {
    const int b    = blockIdx.x;
    const int tid  = threadIdx.x;
    const int wave = tid >> 5;
    const int lane = tid & 31;

    __shared__ float s_cg[245];
    __shared__ float s_rb[80 * 34];          // padded radial basis [edge][r]
    __shared__ float s_sh[80 * 12];          // spherical harmonics [edge][9]
    __shared__ float s_w[80 * CCH];          // per-path radial weights w[e][c]
    __shared__ float s_bufA[6144];           // x0:[12][128] @0 ; x1:[12][128][3] @1536
    __shared__ float s_bufB[6144];
    __shared__ float s_acc[16 * ACW * 5];    // pre-linear accumulator [n][c][k]
    __shared__ float s_y[9 * NA];            // layer-3 used features [feat][atom]
    __shared__ float s_g[9];
    __shared__ float s_H[196];

    for (int i = tid; i < 245; i += 256) s_cg[i] = P.cg[i];
    for (int i = tid; i < 80 * 34; i += 256) s_rb[i] = 0.0f;
    __syncthreads();

    // ---------------- edge features: rb + sh ----------------
    if (tid < NE) {
        int e = tid;
        int i = 0, rem = e;
        while (rem >= 11 - i) { rem -= 11 - i; ++i; }
        int src = i, dst = i + 1 + rem;
        const float* pp = P.pos + ((size_t)b * NA) * 3;
        float dx = pp[dst*3+0] - pp[src*3+0];
        float dy = pp[dst*3+1] - pp[src*3+1];
        float dz = pp[dst*3+2] - pp[src*3+2];
        float r  = sqrtf(dx*dx + dy*dy + dz*dz + 1e-12f);
        float ir = 1.0f / r;
        float ux = dx*ir, uy = dy*ir, uz = dz*ir;
        float tcl = fminf(r, 5.0f);
        float fc  = 0.5f * (cosf(3.14159265358979323846f * tcl * 0.2f) + 1.0f);
        #pragma unroll
        for (int k = 0; k < NR; ++k) {
            float mu = 5.0f * (float)k / 19.0f;
            float d  = r - mu;
            s_rb[e * 34 + k] = expf(-16.0f * d * d) * fc;      // gamma = (20/5)^2
        }
        const float s3  = 1.7320508075688772f;
        const float s15 = 3.8729833462074170f;
        s_sh[e*12+0] = 1.0f;
        s_sh[e*12+1] = s3 * uy;
        s_sh[e*12+2] = s3 * uz;
        s_sh[e*12+3] = s3 * ux;
        s_sh[e*12+4] = s15 * ux * uy;
        s_sh[e*12+5] = s15 * uy * uz;
        s_sh[e*12+6] = 1.1180339887498949f * (3.0f * uz * uz - 1.0f);
        s_sh[e*12+7] = s15 * ux * uz;
        s_sh[e*12+8] = 0.5f * s15 * (ux * ux - uy * uy);
    }
    // ---------------- embedding ----------------
    for (int idx = tid; idx < NA * CCH; idx += 256) {
        int n = idx >> 7, c = idx & 127;
        int z = P.zed[b * NA + n];
        s_bufA[idx] = P.embed[z * CCH + c];
    }
    __syncthreads();

    float* xin  = s_bufA;
    float* xout = s_bufB;

    for (int blk = 0; blk < 9; ++blk) {
        const BlkT B = BLKS[blk];
        const int l3 = B.l3;
        const int K3 = 2 * l3 + 1;

        for (int i = tid; i < 16 * ACW * K3; i += 256) s_acc[i] = 0.0f;
        __syncthreads();

        for (int p = 0; p < B.npaths; ++p) {
            const int l1 = B.p[p].l1, l2 = B.p[p].l2;
            const float* rw = P.rad[B.p[p].rad];

            // ---- radial GEMM: s_w[80x128] = s_rb[80x20] * rw[20x128] (WMMA f32) ----
            {
                const int arow = lane & 15;
                const int kb   = (lane >> 4) * 2;
                for (int t0 = wave; t0 < 40; t0 += 8) {
                    int mt = t0 >> 3, nt = t0 & 7;
                    int Arow = mt * 16 + arow;
                    int ccol = nt * 16 + (lane & 15);
                    v8f acc8 = {0.f,0.f,0.f,0.f,0.f,0.f,0.f,0.f};
                    #pragma unroll
                    for (int ks = 0; ks < 5; ++ks) {
                        int r0 = ks * 4 + kb;
                        v2f a;  a.x  = s_rb[Arow * 34 + r0];     a.y  = s_rb[Arow * 34 + r0 + 1];
                        v2f bb; bb.x = rw[r0 * CCH + ccol];      bb.y = rw[(r0 + 1) * CCH + ccol];
                        acc8 = __builtin_amdgcn_wmma_f32_16x16x4_f32(false, a, false, bb,
                                                                     (short)0, acc8, false, false);
                    }
                    int rbase = mt * 16 + ((lane >> 4) << 3);
                    #pragma unroll
                    for (int v = 0; v < 8; ++v) s_w[(rbase + v) * CCH + ccol] = acc8[v];
                }
            }
            __syncthreads();

            // ---- CG-coupled message + scatter (node-parallel, no atomics) ----
            {
                const int K1 = 2 * l1 + 1, K2 = 2 * l2 + 1;
                const int shoff = l2 * l2;
                const float* cgp = s_cg + cgoff(l1, l2, l3);
                for (int idx = tid; idx < NA * CCH; idx += 256) {
                    int n = idx >> 7, c = idx & 127;
                    float accl[5] = {0.f, 0.f, 0.f, 0.f, 0.f};
                    for (int i = 0; i < n; ++i) {          // in-edges of node n: src i < n
                        int e = ((i * (23 - i)) >> 1) + (n - 1 - i);
                        float wv = s_w[e * CCH + c];
                        const float* xs = (l1 == 0) ? (xin + i * CCH + c)
                                                    : (xin + 1536 + (i * CCH + c) * 3);
                        const float* shp = s_sh + e * 12 + shoff;
                        for (int i1 = 0; i1 < K1; ++i1) {
                            float wx = wv * xs[i1];
                            const float* cgi = cgp + i1 * K2 * K3;
                            for (int j = 0; j < K2; ++j) {
                                float wxs = wx * shp[j];
                                for (int k = 0; k < K3; ++k) accl[k] += wxs * cgi[j * K3 + k];
                            }
                        }
                    }
                    float* ap = s_acc + (n * ACW + c) * K3;
                    for (int k = 0; k < K3; ++k) ap[k] += accl[k];
                }
            }
            __syncthreads();
        }

        if (blk < 6) {
            // ---- channel mixing: out[n][d][k] = (acc[n][:][k]/12) * Wlin (WMMA f32) ----
            const float* lw = P.lin[B.lin];
            const int ntasks = K3 * 8;
            const int arown  = lane & 15;
            const int kb     = (lane >> 4) * 2;
            for (int t0 = wave; t0 < ntasks; t0 += 8) {
                int k = t0 >> 3, nt = t0 & 7;
                int dcol = nt * 16 + (lane & 15);
                v8f acc8 = {0.f,0.f,0.f,0.f,0.f,0.f,0.f,0.f};
                #pragma unroll 4
                for (int ks = 0; ks < 32; ++ks) {
                    int c0 = ks * 4 + kb;
                    v2f a;  a.x  = s_acc[(arown * ACW + c0) * K3 + k];
                            a.y  = s_acc[(arown * ACW + c0 + 1) * K3 + k];
                    v2f bb; bb.x = lw[c0 * CCH + dcol];
                            bb.y = lw[(c0 + 1) * CCH + dcol];
                    acc8 = __builtin_amdgcn_wmma_f32_16x16x4_f32(false, a, false, bb,
                                                                 (short)0, acc8, false, false);
                }
                int rbase = (lane >> 4) << 3;
                #pragma unroll
                for (int v = 0; v < 8; ++v) {
                    int n = rbase + v;
                    if (n < NA) {
                        float val = acc8[v] * (1.0f / 12.0f);
                        int o = (l3 == 0) ? (n * CCH + dcol) : (1536 + (n * CCH + dcol) * 3 + k);
                        if (B.resid) val += xin[o];
                        xout[o] = val;
                    }
                }
            }
            __syncthreads();
        } else {
            // ---- layer-3: only column 0 of Wlin reaches the output (exact DCE) ----
            const float* lw = P.lin[B.lin];
            const int coutN = (l3 == 0) ? 68 : (l3 == 1) ? 36 : 4;
            const int yo    = (l3 == 0) ? 0  : (l3 == 1) ? 1  : 4;
            for (int t0 = tid; t0 < NA * K3; t0 += 256) {
                int n = t0 / K3, k = t0 % K3;
                float s = 0.f;
                for (int c = 0; c < CCH; ++c) s += s_acc[(n * ACW + c) * K3 + k] * lw[c * coutN];
                s_y[(yo + k) * NA + n] = s * (1.0f / 12.0f);
            }
            __syncthreads();
        }

        if (blk == 1 || blk == 3 || blk == 5) { float* tmp = xin; xin = xout; xout = tmp; }
    }

    // ---------------- H build: mean feats -> 14x14 block matrix -> symmetrize ----
    if (tid < 9) {
        float s = 0.f;
        for (int n = 0; n < NA; ++n) s += s_y[tid * NA + n];
        s_g[tid] = s * (1.0f / 12.0f);
    }
    __syncthreads();
    if (tid < 196) {
        int ig = tid / 14, jg = tid % 14;
        const int SL[10] = {0,0,1,0,0,1,0,0,0,0};
        const int SO[10] = {0,1,2,5,6,7,10,11,12,13};
        int li = 0, iloc = 0, lj = 0, jloc = 0;
        for (int s = 0; s < 10; ++s) {
            int sz = 2 * SL[s] + 1;
            if (ig >= SO[s] && ig < SO[s] + sz) { li = SL[s]; iloc = ig - SO[s]; }
            if (jg >= SO[s] && jg < SO[s] + sz) { lj = SL[s]; jloc = jg - SO[s]; }
        }
        float h = 0.f;
        int lmin = (li > lj) ? (li - lj) : (lj - li);
        for (int l3 = lmin; l3 <= li + lj; ++l3) {
            const float* cgp = s_cg + cgoff(li, lj, l3);
            int K2 = 2 * lj + 1, K3n = 2 * l3 + 1;
            const float* g = s_g + ((l3 == 0) ? 0 : (l3 == 1) ? 1 : 4);
            const float* row = cgp + (iloc * K2 + jloc) * K3n;
            for (int k = 0; k < K3n; ++k) h += row[k] * g[k];
        }
        s_H[tid] = h;
    }
    __syncthreads();
    if (tid < 196) {
        int ig = tid / 14, jg = tid % 14;
        P.out[(size_t)b * 196 + tid] = s_H[tid] + s_H[jg * 14 + ig];
    }
}

// ---------------------------------------------------------------------------
// Host launcher: resolve params-dict ordering (insertion vs pytree/alphabetical)
// ---------------------------------------------------------------------------
static const int RAD_POS_INS[23]   = {1,3,5,6,8,9,10,11,13,14,16,17,18,19,21,22,24,25,26,27,29,30,31};
static const int LIN_POS_INS[9]    = {2,4,7,12,15,20,23,28,32};
static const int RAD_POS_ALPHA[23] = {2,3,6,9,7,8,10,11,14,17,15,16,18,19,23,27,24,26,28,30,25,29,31};
static const int LIN_POS_ALPHA[9]  = {0,1,4,5,12,13,20,21,22};

extern "C" void kernel_launch(void* const* d_in, const int* in_sizes, int n_in,
                              void* d_out, int out_size, void* d_ws, size_t ws_size,
                              hipStream_t stream) {
    (void)out_size; (void)ws_size;
    int posIdx = -1, zIdx = -1;
    for (int i = 0; i < n_in; ++i) {
        if (in_sizes[i] == BATCHN * NA * 3 && posIdx < 0) posIdx = i;
        else if (in_sizes[i] == BATCHN * NA && zIdx < 0) zIdx = i;
    }
    if (posIdx < 0) posIdx = 0;
    if (zIdx < 0) zIdx = (posIdx == 1) ? 0 : 1;

    int pidx[64]; int np = 0;
    for (int i = 0; i < n_in && np < 64; ++i)
        if (i != posIdx && i != zIdx) pidx[np++] = i;

    bool ins = (np > 0 && in_sizes[pidx[0]] == 768);   // embed first => insertion order

    KParams P;
    P.pos = (const float*)d_in[posIdx];
    P.zed = (const int*)d_in[zIdx];
    auto pick = [&](int pos) -> const float* {
        if (pos < 0) pos = 0;
        if (pos >= np) pos = (np > 0) ? np - 1 : 0;
        return (const float*)d_in[pidx[pos]];
    };
    P.embed = pick(ins ? 0 : 32);
    for (int r = 0; r < 23; ++r) P.rad[r] = pick(ins ? RAD_POS_INS[r] : RAD_POS_ALPHA[r]);
    for (int l = 0; l < 9; ++l)  P.lin[l] = pick(ins ? LIN_POS_INS[l] : LIN_POS_ALPHA[l]);
    P.cg  = (const float*)d_ws;
    P.out = (float*)d_out;

    so3_cg_init<<<1, 32, 0, stream>>>((float*)d_ws);
    so3_main<<<BATCHN, 256, 0, stream>>>(P);
}